// RESIDUAL_4105988735320
// MI455X (gfx1250) — compile-verified
//
#include <hip/hip_runtime.h>

// ---------------------------------------------------------------------------
// CDNA5 (gfx1250) implementation of the sparse-conv + PFN reference.
// Matrix math uses V_WMMA_F32_16X16X4_F32 (wave32, f32-exact).
// ---------------------------------------------------------------------------

typedef __attribute__((ext_vector_type(2))) float v2f;
typedef __attribute__((ext_vector_type(8))) float v8f;

#define EPS   1e-3f
#define BB    4
#define HH    496
#define WWID  432
#define BHW   (BB * HH * WWID)

__device__ __forceinline__ v8f wmma4(v2f a, v2f b, v8f c) {
  // D = A(16x4,f32) * B(4x16,f32) + C(16x16,f32)
  return __builtin_amdgcn_wmma_f32_16x16x4_f32(
      /*neg_a=*/false, a, /*neg_b=*/false, b,
      /*c_mod=*/(short)0, c, /*reuse_a=*/false, /*reuse_b=*/false);
}

// order-preserving float <-> uint encoding (for atomic max on floats)
__device__ __forceinline__ unsigned fenc(float v) {
  unsigned u = __float_as_uint(v);
  return (u & 0x80000000u) ? ~u : (u | 0x80000000u);
}
__device__ __forceinline__ float fdec(unsigned k) {
  return (k & 0x80000000u) ? __uint_as_float(k & 0x7fffffffu)
                           : __uint_as_float(~k);
}

// ---------------------------------------------------------------------------
// workspace init (runs every call: deterministic)
// ---------------------------------------------------------------------------
__global__ void k_init(int* __restrict__ grid, unsigned* __restrict__ segk,
                       int* __restrict__ flag, float* __restrict__ stats,
                       int n_grid, int n_segk, int n_flag, int n_stats) {
  int i = blockIdx.x * blockDim.x + threadIdx.x;
  if (i < n_grid)  grid[i] = -1;
  if (i < n_segk)  segk[i] = 0x007FFFFFu;   // fenc(-inf)
  if (i < n_flag)  flag[i] = 0;
  if (i < n_stats) stats[i] = 0.f;
}

__global__ void k_scatter(const int* __restrict__ idx, int* __restrict__ grid,
                          int N) {
  int n = blockIdx.x * blockDim.x + threadIdx.x;
  if (n >= N) return;
  int b = idx[n * 3 + 0], h = idx[n * 3 + 1], w = idx[n * 3 + 2];
  grid[(b * HH + h) * WWID + w] = n;
}

__global__ void k_neigh(const int* __restrict__ idx, const int* __restrict__ grid,
                        int* __restrict__ nbr, int N) {
  int n = blockIdx.x * blockDim.x + threadIdx.x;
  if (n >= N) return;
  int b = idx[n * 3 + 0], h = idx[n * 3 + 1], w = idx[n * 3 + 2];
#pragma unroll
  for (int t = 0; t < 9; ++t) {
    int hh = h + t / 3 - 1, ww = w + t % 3 - 1;
    int v = -1;
    if (hh >= 0 && hh < HH && ww >= 0 && ww < WWID)
      v = grid[(b * HH + hh) * WWID + ww];
    nbr[n * 9 + t] = v;
  }
}

// ---------------------------------------------------------------------------
// Submanifold 3x3 conv via gather + WMMA.  4 waves/block, 16 rows/wave.
// Weights (3,3,32,32) HWIO staged in LDS; A tiles staged per wave in LDS.
// Fused: +bias, per-channel sum/sumsq accumulation for BatchNorm.
// ---------------------------------------------------------------------------
__global__ __launch_bounds__(128) void k_conv(
    const float* __restrict__ fin, const int* __restrict__ nbr,
    const float* __restrict__ wgt, const float* __restrict__ bias,
    float* __restrict__ fout, float* __restrict__ stat, int N) {
  __shared__ __align__(16) float lw[9 * 32 * 32];
  __shared__ __align__(16) float la[4][16 * 32];

  const int tid = threadIdx.x;
  for (int i = tid; i < 9 * 32 * 32; i += 128) lw[i] = wgt[i];
  __syncthreads();

  const int wave = tid >> 5, lane = tid & 31;
  const int half = lane >> 4, l16 = lane & 15;
  const int row0 = (blockIdx.x * 4 + wave) * 16;

  v8f acc0 = {}; v8f acc1 = {};

  for (int t = 0; t < 9; ++t) {
    // stage gathered A tile (16 rows x 32 ch): lane -> (row=lane>>1, 16-ch half)
    const int r = lane >> 1, cb = (lane & 1) * 16;
    const int grow = row0 + r;
    const int nid = (grow < N) ? nbr[grow * 9 + t] : -1;
    float4 p0 = {0.f, 0.f, 0.f, 0.f}, p1 = p0, p2 = p0, p3 = p0;
    if (nid >= 0) {
      const float4* s = (const float4*)(fin + nid * 32 + cb);
      p0 = s[0]; p1 = s[1]; p2 = s[2]; p3 = s[3];
    }
    float4* d = (float4*)&la[wave][r * 32 + cb];
    d[0] = p0; d[1] = p1; d[2] = p2; d[3] = p3;
    __syncthreads();

    const float* A  = la[wave];
    const float* Wt = lw + t * 1024;   // Wt[ci*32 + co]
#pragma unroll
    for (int k = 0; k < 8; ++k) {
      const int k0 = k * 4 + 2 * half;
      v2f a;  a.x  = A[l16 * 32 + k0];        a.y  = A[l16 * 32 + k0 + 1];
      v2f b0; b0.x = Wt[k0 * 32 + l16];       b0.y = Wt[(k0 + 1) * 32 + l16];
      v2f b1; b1.x = Wt[k0 * 32 + 16 + l16];  b1.y = Wt[(k0 + 1) * 32 + 16 + l16];
      acc0 = wmma4(a, b0, acc0);
      acc1 = wmma4(a, b1, acc1);
    }
    __syncthreads();
  }

  const float bc0 = bias[l16], bc1 = bias[16 + l16];
  float s0 = 0.f, q0 = 0.f, s1 = 0.f, q1 = 0.f;
#pragma unroll
  for (int r = 0; r < 8; ++r) {
    const int row = row0 + r + 8 * half;   // C/D layout: VGPR r holds M=r / r+8
    if (row < N) {
      float v0 = acc0[r] + bc0;
      float v1 = acc1[r] + bc1;
      fout[row * 32 + l16]      = v0;
      fout[row * 32 + 16 + l16] = v1;
      s0 += v0; q0 += v0 * v0; s1 += v1; q1 += v1 * v1;
    }
  }
  atomicAdd(&stat[l16],      s0);
  atomicAdd(&stat[32 + l16], q0);
  atomicAdd(&stat[16 + l16], s1);
  atomicAdd(&stat[48 + l16], q1);
}

// ---------------------------------------------------------------------------
// BatchNorm apply kernels (training-mode biased stats)
// ---------------------------------------------------------------------------
__global__ void k_bn_relu(const float* __restrict__ y, const float* __restrict__ stat,
                          const float* __restrict__ g, const float* __restrict__ be,
                          float* __restrict__ out, int rows, float inv) {
  int i = blockIdx.x * blockDim.x + threadIdx.x;
  if (i >= rows * 32) return;
  int c = i & 31;
  float mean = stat[c] * inv;
  float var  = stat[32 + c] * inv - mean * mean;
  float v = (y[i] - mean) * rsqrtf(var + EPS) * g[c] + be[c];
  out[i] = fmaxf(v, 0.f);
}

__global__ void k_bn_add_relu(const float* __restrict__ y, const float* __restrict__ stat,
                              const float* __restrict__ g, const float* __restrict__ be,
                              const float* __restrict__ res, float* __restrict__ out,
                              int rows, float inv) {
  int i = blockIdx.x * blockDim.x + threadIdx.x;
  if (i >= rows * 32) return;
  int c = i & 31;
  float mean = stat[c] * inv;
  float var  = stat[32 + c] * inv - mean * mean;
  float v = (y[i] - mean) * rsqrtf(var + EPS) * g[c] + be[c];
  out[i] = fmaxf(v + res[i], 0.f);
}

// ---------------------------------------------------------------------------
// PFN GEMMs (WMMA) with fused BN-stats accumulation
// ---------------------------------------------------------------------------
__global__ __launch_bounds__(128) void k_gemm8(
    const float* __restrict__ A, const float* __restrict__ Bw,
    float* __restrict__ out, float* __restrict__ stat, int M) {
  const int tid = threadIdx.x, wave = tid >> 5, lane = tid & 31;
  const int half = lane >> 4, l16 = lane & 15;
  const int row0 = (blockIdx.x * 4 + wave) * 16;
  const int arow = row0 + l16;
  v8f acc0 = {}; v8f acc1 = {};
#pragma unroll
  for (int k = 0; k < 2; ++k) {
    const int k0 = k * 4 + 2 * half;
    v2f a;
    if (arow < M) { a.x = A[arow * 8 + k0]; a.y = A[arow * 8 + k0 + 1]; }
    else          { a.x = 0.f;              a.y = 0.f; }
    v2f b0, b1;
    b0.x = Bw[k0 * 32 + l16];       b0.y = Bw[(k0 + 1) * 32 + l16];
    b1.x = Bw[k0 * 32 + 16 + l16];  b1.y = Bw[(k0 + 1) * 32 + 16 + l16];
    acc0 = wmma4(a, b0, acc0);
    acc1 = wmma4(a, b1, acc1);
  }
  float s0 = 0.f, q0 = 0.f, s1 = 0.f, q1 = 0.f;
#pragma unroll
  for (int r = 0; r < 8; ++r) {
    const int row = row0 + r + 8 * half;
    if (row < M) {
      float v0 = acc0[r], v1 = acc1[r];
      out[row * 32 + l16]      = v0;
      out[row * 32 + 16 + l16] = v1;
      s0 += v0; q0 += v0 * v0; s1 += v1; q1 += v1 * v1;
    }
  }
  atomicAdd(&stat[l16],      s0);
  atomicAdd(&stat[32 + l16], q0);
  atomicAdd(&stat[16 + l16], s1);
  atomicAdd(&stat[48 + l16], q1);
}

// inputs = relu(bn(opf_lin)) + feat2[unq_inv]  (in place on buf)
__global__ void k_pfn_mix(float* __restrict__ buf, const float* __restrict__ stat,
                          const float* __restrict__ g, const float* __restrict__ be,
                          const float* __restrict__ feat2, const int* __restrict__ unq,
                          int M, float inv) {
  int i = blockIdx.x * blockDim.x + threadIdx.x;
  if (i >= M * 32) return;
  int c = i & 31, m = i >> 5;
  float mean = stat[c] * inv;
  float var  = stat[32 + c] * inv - mean * mean;
  float v = fmaxf((buf[i] - mean) * rsqrtf(var + EPS) * g[c] + be[c], 0.f);
  buf[i] = v + feat2[unq[m] * 32 + c];
}

// in-place M x 32 @ 32 x 32 (per-tile read-then-write is safe)
__global__ __launch_bounds__(128) void k_gemm32(
    float* __restrict__ buf, const float* __restrict__ Bw,
    float* __restrict__ stat, int M) {
  const int tid = threadIdx.x, wave = tid >> 5, lane = tid & 31;
  const int half = lane >> 4, l16 = lane & 15;
  const int row0 = (blockIdx.x * 4 + wave) * 16;
  const int arow = row0 + l16;
  v8f acc0 = {}; v8f acc1 = {};
#pragma unroll
  for (int k = 0; k < 8; ++k) {
    const int k0 = k * 4 + 2 * half;
    v2f a;
    if (arow < M) { a.x = buf[arow * 32 + k0]; a.y = buf[arow * 32 + k0 + 1]; }
    else          { a.x = 0.f;                 a.y = 0.f; }
    v2f b0, b1;
    b0.x = Bw[k0 * 32 + l16];       b0.y = Bw[(k0 + 1) * 32 + l16];
    b1.x = Bw[k0 * 32 + 16 + l16];  b1.y = Bw[(k0 + 1) * 32 + 16 + l16];
    acc0 = wmma4(a, b0, acc0);
    acc1 = wmma4(a, b1, acc1);
  }
  float s0 = 0.f, q0 = 0.f, s1 = 0.f, q1 = 0.f;
#pragma unroll
  for (int r = 0; r < 8; ++r) {
    const int row = row0 + r + 8 * half;
    if (row < M) {
      float v0 = acc0[r], v1 = acc1[r];
      buf[row * 32 + l16]      = v0;
      buf[row * 32 + 16 + l16] = v1;
      s0 += v0; q0 += v0 * v0; s1 += v1; q1 += v1 * v1;
    }
  }
  atomicAdd(&stat[l16],      s0);
  atomicAdd(&stat[32 + l16], q0);
  atomicAdd(&stat[16 + l16], s1);
  atomicAdd(&stat[48 + l16], q1);
}

// xr = relu(bn(buf)); segment max + present flag via atomics
__global__ void k_segmax(const float* __restrict__ buf, const float* __restrict__ stat,
                         const float* __restrict__ g, const float* __restrict__ be,
                         const int* __restrict__ unq, unsigned* __restrict__ segk,
                         int* __restrict__ flag, int M, float inv) {
  int i = blockIdx.x * blockDim.x + threadIdx.x;
  if (i >= M * 32) return;
  int c = i & 31, m = i >> 5;
  float mean = stat[c] * inv;
  float var  = stat[32 + c] * inv - mean * mean;
  float v = fmaxf((buf[i] - mean) * rsqrtf(var + EPS) * g[c] + be[c], 0.f);
  int seg = unq[m];
  atomicMax(&segk[seg * 32 + c], fenc(v));
  if (c == 0) flag[seg] = 1;
}

__global__ void k_final(const unsigned* __restrict__ segk, const int* __restrict__ flag,
                        const float* __restrict__ feat2, float* __restrict__ out, int N) {
  int i = blockIdx.x * blockDim.x + threadIdx.x;
  if (i >= N * 32) return;
  int n = i >> 5;
  out[i] = flag[n] ? fdec(segk[i]) : feat2[i];
}

// ---------------------------------------------------------------------------
// host orchestration
// ---------------------------------------------------------------------------
extern "C" void kernel_launch(void* const* d_in, const int* in_sizes, int n_in,
                              void* d_out, int out_size, void* d_ws, size_t ws_size,
                              hipStream_t stream) {
  (void)n_in; (void)out_size; (void)ws_size;
  const float* x    = (const float*)d_in[0];
  const int*   idx  = (const int*)d_in[1];
  const float* opf  = (const float*)d_in[2];
  const int*   unq  = (const int*)d_in[3];
  const float* w1a  = (const float*)d_in[5];
  const float* w1b  = (const float*)d_in[6];
  const float* w2a  = (const float*)d_in[7];
  const float* w2b  = (const float*)d_in[8];
  const float* b1a  = (const float*)d_in[9];
  const float* b1b  = (const float*)d_in[10];
  const float* b2a  = (const float*)d_in[11];
  const float* b2b  = (const float*)d_in[12];
  const float* g1a  = (const float*)d_in[13];
  const float* g1b  = (const float*)d_in[14];
  const float* g2a  = (const float*)d_in[15];
  const float* g2b  = (const float*)d_in[16];
  const float* cg   = (const float*)d_in[17];
  const float* lg   = (const float*)d_in[18];
  const float* be1a = (const float*)d_in[19];
  const float* be1b = (const float*)d_in[20];
  const float* be2a = (const float*)d_in[21];
  const float* be2b = (const float*)d_in[22];
  const float* cb   = (const float*)d_in[23];
  const float* lb   = (const float*)d_in[24];
  const float* cw   = (const float*)d_in[25];
  const float* lw   = (const float*)d_in[26];

  const int N = in_sizes[0] / 32;
  const int M = in_sizes[2] / 8;

  auto alignup = [](size_t v) { return (v + 255) & ~(size_t)255; };
  char* p = (char*)d_ws;
  int*      grid = (int*)p;      p += alignup((size_t)BHW * 4);
  int*      nbr  = (int*)p;      p += alignup((size_t)N * 9 * 4);
  float*    bufA = (float*)p;    p += alignup((size_t)N * 32 * 4);
  float*    bufB = (float*)p;    p += alignup((size_t)N * 32 * 4);
  float*    bufC = (float*)p;    p += alignup((size_t)N * 32 * 4);
  float*    bufM = (float*)p;    p += alignup((size_t)M * 32 * 4);
  unsigned* segk = (unsigned*)p; p += alignup((size_t)N * 32 * 4);
  int*      flag = (int*)p;      p += alignup((size_t)N * 4);
  float*    stats = (float*)p;   p += alignup((size_t)6 * 64 * 4);
  float*    out  = (float*)d_out;

  const int T = 256;
  int initCount = BHW > N * 32 ? BHW : N * 32;
  k_init<<<(initCount + T - 1) / T, T, 0, stream>>>(grid, segk, flag, stats,
                                                    BHW, N * 32, N, 6 * 64);
  k_scatter<<<(N + T - 1) / T, T, 0, stream>>>(idx, grid, N);
  k_neigh<<<(N + T - 1) / T, T, 0, stream>>>(idx, grid, nbr, N);

  const int   cBlk = (N + 63) / 64;
  const int   eN   = (N * 32 + T - 1) / T;
  const float invN = 1.f / (float)N;
  const float invM = 1.f / (float)M;
  float *s0 = stats, *s1 = stats + 64, *s2 = stats + 128,
        *s3 = stats + 192, *s4 = stats + 256, *s5 = stats + 320;

  // block 1
  k_conv<<<cBlk, 128, 0, stream>>>(x, nbr, w1a, b1a, bufA, s0, N);
  k_bn_relu<<<eN, T, 0, stream>>>(bufA, s0, g1a, be1a, bufB, N, invN);
  k_conv<<<cBlk, 128, 0, stream>>>(bufB, nbr, w1b, b1b, bufA, s1, N);
  k_bn_add_relu<<<eN, T, 0, stream>>>(bufA, s1, g1b, be1b, x, bufC, N, invN);
  // block 2
  k_conv<<<cBlk, 128, 0, stream>>>(bufC, nbr, w2a, b2a, bufA, s2, N);
  k_bn_relu<<<eN, T, 0, stream>>>(bufA, s2, g2a, be2a, bufB, N, invN);
  k_conv<<<cBlk, 128, 0, stream>>>(bufB, nbr, w2b, b2b, bufA, s3, N);
  k_bn_add_relu<<<eN, T, 0, stream>>>(bufA, s3, g2b, be2b, bufC, bufB, N, invN);
  // bufB now holds feat2 (N x 32)

  // PFN
  const int gBlk = (M + 63) / 64;
  const int eM   = (M * 32 + T - 1) / T;
  k_gemm8<<<gBlk, 128, 0, stream>>>(opf, cw, bufM, s4, M);
  k_pfn_mix<<<eM, T, 0, stream>>>(bufM, s4, cg, cb, bufB, unq, M, invM);
  k_gemm32<<<gBlk, 128, 0, stream>>>(bufM, lw, s5, M);
  k_segmax<<<eM, T, 0, stream>>>(bufM, s5, lg, lb, unq, segk, flag, M, invM);
  k_final<<<eN, T, 0, stream>>>(segk, flag, bufB, out, N);
}